// WarpKptAdvanced_60241211294087
// MI455X (gfx1250) — compile-verified
//
#include <hip/hip_runtime.h>
#include <math.h>

typedef float v2f __attribute__((ext_vector_type(2)));
typedef float v8f __attribute__((ext_vector_type(8)));

#define KPTS    512
#define CHUNKS  (KPTS / 16)
#define SSTRIDE 516                       // score row stride (floats); 8*516%64=32 -> conflict-free scatter
#define RSTRIDE 17                        // runs row stride (u32)
#define LSTRIDE 33                        // selres row stride (u32)

// dynamic-LDS byte offsets
#define OFF_KQ   0                        // float4[512]  {-2kx,-2ky,-2kz,|k|^2}   8192 B
#define OFF_DK   8192                     // float4[512]  {canon-k, pad}           8192 B
#define OFF_R2   16384                    // float4[512]  {rx^2,ry^2,rz^2, pad}    8192 B
#define OFF_SC   24576                    // f32 [16][516] score matrix           33024 B
#define OFF_RUNS 57600                    // u32 [32][17]  sorted runs             2176 B
#define OFF_SEL  59776                    // u32 [16][33]  per-row top-32 keys     2112 B
#define SMEM_BYTES 61888

__device__ __forceinline__ unsigned ord32(float f) {
    unsigned u = __float_as_uint(f);
    return (u & 0x80000000u) ? ~u : (u | 0x80000000u);
}

// One wave per block; 16 points per block (one 16x4 f32 WMMA A-tile).
__global__ void __launch_bounds__(32)
warp_kpt_advanced_kernel(const float* __restrict__ pts,
                         const float* __restrict__ transform,
                         const float* __restrict__ kpt3d,
                         const float* __restrict__ canon,
                         const float* __restrict__ radius,
                         const int*   __restrict__ tptr,
                         float*       __restrict__ out,
                         int N)
{
    extern __shared__ char smem[];
    float4*   kq    = (float4*)  (smem + OFF_KQ);
    float4*   dk4   = (float4*)  (smem + OFF_DK);
    float4*   r24   = (float4*)  (smem + OFF_R2);
    float*    score = (float*)   (smem + OFF_SC);
    unsigned* runs  = (unsigned*)(smem + OFF_RUNS);
    unsigned* sel   = (unsigned*)(smem + OFF_SEL);

    const int  lane = threadIdx.x;        // 0..31
    const int  s0   = lane & 15;          // point / column slot within tile
    const bool lo   = lane < 16;
    const int  t    = tptr[0];
    const float* kp = kpt3d + (size_t)t * KPTS * 3;

    // ---- Phase 0: keypoint tables into LDS ----
    for (int n = lane; n < KPTS; n += 32) {
        float kx = kp[n*3+0], ky = kp[n*3+1], kz = kp[n*3+2];
        kq[n] = make_float4(-2.0f*kx, -2.0f*ky, -2.0f*kz, kx*kx + ky*ky + kz*kz);
        float cx = canon[n*3+0], cy = canon[n*3+1], cz = canon[n*3+2];
        dk4[n] = make_float4(cx - kx, cy - ky, cz - kz, 0.0f);
        float rx = radius[n*3+0], ry = radius[n*3+1], rz = radius[n*3+2];
        r24[n] = make_float4(rx*rx, ry*ry, rz*rz, 0.0f);
    }

    // ---- own point (lanes l and l+16 both hold point s0) ----
    const float* Tm = transform + (size_t)t * 12;
    float T00=Tm[0], T01=Tm[1], T02=Tm[2],  T03=Tm[3];
    float T10=Tm[4], T11=Tm[5], T12=Tm[6],  T13=Tm[7];
    float T20=Tm[8], T21=Tm[9], T22=Tm[10], T23=Tm[11];

    const int q = blockIdx.x * 16 + s0;   // N is a multiple of 16
    float x = pts[q*3+0], y = pts[q*3+1], z = pts[q*3+2];
    float px = T00*x + T01*y + T02*z + T03;
    float py = T10*x + T11*y + T12*z + T13;
    float pz = T20*x + T21*y + T22*z + T23;
    float psq = px*px + py*py + pz*pz;

    __syncthreads();                      // single-wave WG: NOP barrier / LDS fence

    // ---- Phase 1: WMMA distance engine ----
    // A (16x4 f32): lane<16 -> {px,py} (K0,K1); lane>=16 -> {pz,1} (K2,K3)
    v2f A;  A[0] = lo ? px : pz;  A[1] = lo ? py : 1.0f;

    #pragma unroll 2
    for (int j = 0; j < CHUNKS; ++j) {
        float4 kv = kq[j*16 + s0];        // lanes l / l+16 broadcast same address
        v2f B;  B[0] = lo ? kv.x : kv.z;  B[1] = lo ? kv.y : kv.w;
        v8f C = {};
        C = __builtin_amdgcn_wmma_f32_16x16x4_f32(false, A, false, B,
                                                  (short)0, C, false, false);
        const int rbase = lo ? 0 : 8;     // C row striping
        #pragma unroll
        for (int r = 0; r < 8; ++r)
            score[(rbase + r)*SSTRIDE + j*16 + s0] = C[r];  // |k|^2 - 2 p.k
    }
    __syncthreads();

    // ---- Phase 2: cooperative exact top-32 per row (uniform control flow) ----
    #pragma unroll 1
    for (int m = 0; m < 16; ++m) {
        // each lane grabs 16 strided candidates of row m, packs sortable keys
        unsigned keys[16];
        #pragma unroll
        for (int i = 0; i < 16; ++i) {
            float sc = score[m*SSTRIDE + i*32 + lane];
            keys[i] = (ord32(sc) & 0xFFFFFE00u) | (unsigned)(i*32 + lane);
        }
        // bitonic sort 16 in registers (static indices, branchless)
        #pragma unroll
        for (int k = 2; k <= 16; k <<= 1) {
            #pragma unroll
            for (int j = k >> 1; j > 0; j >>= 1) {
                #pragma unroll
                for (int i = 0; i < 16; ++i) {
                    int ixj = i ^ j;
                    if (ixj > i) {
                        unsigned a = keys[i], b = keys[ixj];
                        unsigned mn = min(a, b), mx = max(a, b);
                        if ((i & k) == 0) { keys[i] = mn; keys[ixj] = mx; }
                        else              { keys[i] = mx; keys[ixj] = mn; }
                    }
                }
            }
        }
        // spill sorted run so the head pointer can be advanced dynamically
        #pragma unroll
        for (int i = 0; i < 16; ++i) runs[lane*RSTRIDE + i] = keys[i];

        // 32 rounds of wave-wide argmin over run heads; lane rr keeps round rr
        unsigned mykey = 0xFFFFFFFFu;
        unsigned head  = runs[lane*RSTRIDE + 0];
        int hidx = 0;
        #pragma unroll 1
        for (int rr = 0; rr < 32; ++rr) {
            unsigned mn = head;
            #pragma unroll
            for (int s = 16; s > 0; s >>= 1)
                mn = min(mn, (unsigned)__shfl_xor((int)mn, s));
            if (lane == rr) mykey = mn;
            if (head == mn) {             // keys globally unique -> exactly one lane
                ++hidx;
                head = (hidx < 16) ? runs[lane*RSTRIDE + hidx] : 0xFFFFFFFFu;
            }
        }
        sel[m*LSTRIDE + lane] = mykey;    // lane rr holds rr-th nearest of row m
        __syncthreads();
    }
    __syncthreads();

    // ---- Phase 3: RBF blend; 2 lanes per point, 16 survivors each ----
    const int half = lane >> 4;
    float numx = 0.f, numy = 0.f, numz = 0.f;
    float denx = 0.f, deny = 0.f, denz = 0.f;
    #pragma unroll 4
    for (int i = 0; i < 16; ++i) {
        unsigned key = sel[s0*LSTRIDE + half*16 + i];
        int   col = (int)(key & 0x1FFu);
        float sc  = score[s0*SSTRIDE + col];            // exact f32 score
        float d2  = fmaxf(sc + psq, 0.0f);              // d^2 = |p|^2 + |k|^2 - 2p.k
        float4 r2 = r24[col];
        float4 dk = dk4[col];
        float wx = fmaxf(__expf(-d2 * r2.x), 1e-10f);
        float wy = fmaxf(__expf(-d2 * r2.y), 1e-10f);
        float wz = fmaxf(__expf(-d2 * r2.z), 1e-10f);
        numx += dk.x * wx;  numy += dk.y * wy;  numz += dk.z * wz;
        denx += wx;         deny += wy;         denz += wz;
    }
    // combine the two half-lanes of each point
    numx += __shfl_xor(numx, 16);  numy += __shfl_xor(numy, 16);
    numz += __shfl_xor(numz, 16);
    denx += __shfl_xor(denx, 16);  deny += __shfl_xor(deny, 16);
    denz += __shfl_xor(denz, 16);

    if (lo) {
        out[q*3+0] = px + numx / denx;
        out[q*3+1] = py + numy / deny;
        out[q*3+2] = pz + numz / denz;
    }
}

extern "C" void kernel_launch(void* const* d_in, const int* in_sizes, int n_in,
                              void* d_out, int out_size, void* d_ws, size_t ws_size,
                              hipStream_t stream) {
    const float* pts       = (const float*)d_in[0];   // [N,3]
    const float* transform = (const float*)d_in[1];   // [T,3,4]
    const float* kpt3d     = (const float*)d_in[2];   // [T,512,3]
    const float* canon     = (const float*)d_in[3];   // [1,512,3]
    const float* radius    = (const float*)d_in[4];   // [1,512,3]
    const int*   t         = (const int*)  d_in[5];   // scalar
    float*       out       = (float*)d_out;           // [N,3]

    const int N = in_sizes[0] / 3;                    // 131072
    const int blocks = N / 16;                        // 8192
    hipLaunchKernelGGL(warp_kpt_advanced_kernel, dim3(blocks), dim3(32),
                       SMEM_BYTES, stream,
                       pts, transform, kpt3d, canon, radius, t, out, N);
}